// ComplementarityScoreHead_82025285419559
// MI455X (gfx1250) — compile-verified
//
#include <hip/hip_runtime.h>
#include <stdint.h>

typedef __attribute__((ext_vector_type(2))) float v2f;
typedef __attribute__((ext_vector_type(4))) float v4f;
typedef __attribute__((ext_vector_type(8))) float v8f;

#define N_NODES 8192
#define F_IN    128
#define H_DIM   256
#define N_EDGES 262144

// LDS layout: K-pair interleaved. Element (k, col) lives at
//   Bs[(k>>1)*PSTRIDE + col*2 + (k&1)]
// PSTRIDE=160 floats: khalf=1 lanes sit +32 banks from khalf=0 lanes, so each
// ds_load_b64 (32 lanes x 2 banks) covers all 64 banks exactly once.
#define PSTRIDE 160

// C[M,N] = (A[M,K] @ B[K,N]) + bias[N], optional ReLU.
// Workgroup (8 waves) computes a 128x64 strip. The K x 64 slab of B is staged
// once into LDS with CDNA5 async global->LDS scatter copies (ASYNCcnt), then
// each wave runs fp32 WMMA 16x16x4; every B fragment is one aligned ds_load_b64.
__global__ __launch_bounds__(256) void gemm_bias_wmma_kernel(
    const float* __restrict__ A, const float* __restrict__ B,
    const float* __restrict__ bias, float* __restrict__ C,
    int M, int K, int N, int relu)
{
    __shared__ float Bs[(H_DIM / 2) * PSTRIDE];      // max K=256 -> 80 KB

    const int tilesX = N >> 6;                       // 64-col strips
    const int bm0 = (blockIdx.x / tilesX) * 128;     // block row base
    const int n0  = (blockIdx.x % tilesX) * 64;      // block col base

    // ---- Stage B[0..K-1][n0..n0+63] into interleaved LDS via async b32 scatter ----
    // thread t owns column (t&63) of rows k = (t>>6), (t>>6)+4, ... (coalesced 128B/wave)
    {
        const int k0  = threadIdx.x >> 6;            // 0..3
        const int col = threadIdx.x & 63;
        const float* g = B + (size_t)k0 * N + n0 + col;
        unsigned lds = (unsigned)(uintptr_t)(&Bs[(k0 >> 1) * PSTRIDE + col * 2 + (k0 & 1)]);
        for (int k = k0; k < K; k += 4) {
            asm volatile("global_load_async_to_lds_b32 %0, %1, off"
                         :: "v"(lds), "v"(g) : "memory");
            g   += (size_t)4 * N;
            lds += 2 * PSTRIDE * 4;                  // two pair-rows, in bytes
        }
    }
    asm volatile("s_wait_asynccnt 0x0" ::: "memory");
    __syncthreads();

    const int lane  = threadIdx.x & 31;
    const int wave  = threadIdx.x >> 5;
    const int row16 = lane & 15;   // M index in A-frag / N index in B,C,D frags
    const int khalf = lane >> 4;   // selects K pair {0,1} vs {2,3}

    const int m0 = bm0 + wave * 16;

    v8f acc0 = {}, acc1 = {}, acc2 = {}, acc3 = {};

    // A-frag: lane holds A[m0+row16][k + 2*khalf + {0,1}] -> one contiguous b64 load
    const float* arow = A + (size_t)(m0 + row16) * K + 2 * khalf;
    // B-frag: {B[k+2*khalf][col], B[k+2*khalf+1][col]} is one aligned b64 in LDS
    const float* bbase = &Bs[khalf * PSTRIDE + row16 * 2];

    #pragma unroll 2
    for (int k = 0; k < K; k += 4) {
        v2f a = *(const v2f*)(arow + k);
        const float* bk = bbase + (k >> 1) * PSTRIDE;
        v2f b0 = *(const v2f*)(bk + 0);    // j=0: col base + 0
        v2f b1 = *(const v2f*)(bk + 32);   // j=1: +16 cols = +32 floats = offset 128B
        v2f b2 = *(const v2f*)(bk + 64);   // j=2
        v2f b3 = *(const v2f*)(bk + 96);   // j=3
        acc0 = __builtin_amdgcn_wmma_f32_16x16x4_f32(false, a, false, b0, (short)0, acc0, false, false);
        acc1 = __builtin_amdgcn_wmma_f32_16x16x4_f32(false, a, false, b1, (short)0, acc1, false, false);
        acc2 = __builtin_amdgcn_wmma_f32_16x16x4_f32(false, a, false, b2, (short)0, acc2, false, false);
        acc3 = __builtin_amdgcn_wmma_f32_16x16x4_f32(false, a, false, b3, (short)0, acc3, false, false);
    }

    // C/D frag: VGPR r holds row (m0 + r + 8*khalf), col (n0 + j*16 + row16)
    v8f accs[4] = {acc0, acc1, acc2, acc3};
    const int rbase = m0 + 8 * khalf;
    #pragma unroll
    for (int j = 0; j < 4; ++j) {
        const int col = n0 + j * 16 + row16;
        const float bs = bias[col];
        #pragma unroll
        for (int r = 0; r < 8; ++r) {
            float v = accs[j][r] + bs;
            if (relu) v = fmaxf(v, 0.0f);
            C[(size_t)(rbase + r) * N + col] = v;
        }
    }
}

// In-place row L2-normalize (256 cols), one wave per row; eps semantics match torch.
__global__ __launch_bounds__(256) void l2norm_rows_kernel(float* __restrict__ h)
{
    const int lane = threadIdx.x & 31;
    const int wave = threadIdx.x >> 5;
    const int row  = blockIdx.x * 8 + wave;

    v4f* p = (v4f*)(h + (size_t)row * H_DIM);   // 64 x v4f per row
    v4f a = p[lane];
    v4f b = p[lane + 32];
    float ss = a.x*a.x + a.y*a.y + a.z*a.z + a.w*a.w
             + b.x*b.x + b.y*b.y + b.z*b.z + b.w*b.w;
    #pragma unroll
    for (int m = 16; m; m >>= 1) ss += __shfl_xor(ss, m, 32);
    const float scale = 1.0f / fmaxf(sqrtf(ss), 1e-12f);
    p[lane]      = a * scale;
    p[lane + 32] = b * scale;
}

// Zero-fill the 256 MB output with non-temporal 128-bit streaming stores.
__global__ __launch_bounds__(256) void zero_out_kernel(float* __restrict__ out, size_t n4)
{
    v4f z = {};
    v4f* o = (v4f*)out;
    size_t i = (size_t)blockIdx.x * 256 + threadIdx.x;
    const size_t stride = (size_t)gridDim.x * 256;
    for (; i < n4; i += stride)
        __builtin_nontemporal_store(z, o + i);
}

// One wave per edge: coalesced 1 KB row reads (L2-resident), wave reduction, scatter 4B store.
__global__ __launch_bounds__(256) void edge_corr_kernel(
    const int* __restrict__ ei, const float* __restrict__ hn, float* __restrict__ out)
{
    const int lane = threadIdx.x & 31;
    const int wave = threadIdx.x >> 5;
    const int e    = blockIdx.x * 8 + wave;

    const int src = ei[e];
    const int dst = ei[N_EDGES + e];

    const v4f* pa = (const v4f*)(hn + (size_t)src * H_DIM);
    const v4f* pb = (const v4f*)(hn + (size_t)dst * H_DIM);
    v4f a0 = pa[lane], a1 = pa[lane + 32];
    v4f b0 = pb[lane], b1 = pb[lane + 32];
    float s = a0.x*b0.x + a0.y*b0.y + a0.z*b0.z + a0.w*b0.w
            + a1.x*b1.x + a1.y*b1.y + a1.z*b1.z + a1.w*b1.w;
    #pragma unroll
    for (int m = 16; m; m >>= 1) s += __shfl_xor(s, m, 32);
    if (lane == 0)
        out[(size_t)src * N_NODES + dst] = s;
}

extern "C" void kernel_launch(void* const* d_in, const int* in_sizes, int n_in,
                              void* d_out, int out_size, void* d_ws, size_t ws_size,
                              hipStream_t stream)
{
    const float* x  = (const float*)d_in[0];
    const int*   ei = (const int*)  d_in[1];
    const float* W1 = (const float*)d_in[2];
    const float* b1 = (const float*)d_in[3];
    const float* W2 = (const float*)d_in[4];
    const float* b2 = (const float*)d_in[5];
    float* out = (float*)d_out;

    float* t = (float*)d_ws;                       // [8192,256] relu(x@W1+b1)
    float* h = t + (size_t)N_NODES * H_DIM;        // [8192,256] h, normalized in place

    const int gemm_blocks = (N_NODES / 128) * (H_DIM / 64);   // 64 * 4 = 256 blocks

    // h = relu(x @ W1 + b1) @ W2 + b2
    gemm_bias_wmma_kernel<<<gemm_blocks, 256, 0, stream>>>(x, W1, b1, t, N_NODES, F_IN, H_DIM, 1);
    gemm_bias_wmma_kernel<<<gemm_blocks, 256, 0, stream>>>(t, W2, b2, h, N_NODES, H_DIM, H_DIM, 0);
    // h_norm rows
    l2norm_rows_kernel<<<N_NODES / 8, 256, 0, stream>>>(h);
    // out = 0 (the ~11 us bandwidth floor), then scatter edge correlations
    zero_out_kernel<<<2048, 256, 0, stream>>>(out, (size_t)N_NODES * N_NODES / 4);
    edge_corr_kernel<<<N_EDGES / 8, 256, 0, stream>>>(ei, h, out);
}